// BIDAF_76922864271822
// MI455X (gfx1250) — compile-verified
//
#include <hip/hip_runtime.h>

typedef __attribute__((ext_vector_type(2))) float v2f;
typedef __attribute__((ext_vector_type(8))) float v8f;

constexpr int NB = 64, NT = 1024, NJ = 128, ND = 256;
constexpr int TB = 128;                    // T rows per block in k1
constexpr int KC = 64;                     // d-chunk width
constexpr int US_STRIDE = 68;              // padded LDS stride for U/H chunks
constexpr int AS_STRIDE = 132;             // padded LDS stride for A tile
constexpr int US_OFF = 0;
constexpr int HS_OFF = NJ * US_STRIDE;     // 8704
constexpr int AS_OFF = HS_OFF;             // A tiles overlay H chunk (phase 2)
constexpr int AS_WAVE = 16 * AS_STRIDE;    // 2112 floats per wave
constexpr int SU_OFF = AS_OFF + 8 * AS_WAVE;   // 25600
constexpr int LDS_FLOATS = SU_OFF + NJ;        // 25728 floats = 102912 B

__device__ inline v8f zero8() {
  v8f z;
#pragma unroll
  for (int i = 0; i < 8; ++i) z[i] = 0.0f;
  return z;
}

// ---------------- k0: sh[b,t] = H . w_h ; su[b,j] = U . w_u ----------------
__global__ __launch_bounds__(256) void k0_rowdots(
    const float* __restrict__ H, const float* __restrict__ U,
    const float* __restrict__ w_h, const float* __restrict__ w_u,
    float* __restrict__ sh, float* __restrict__ su) {
  const int wave = threadIdx.x >> 5;
  const int lane = threadIdx.x & 31;
  long row = (long)blockIdx.x * 8 + wave;
  const float* src; const float* w; float* dst;
  if (row < (long)NB * NT) {
    src = H + row * ND; w = w_h; dst = sh + row;
  } else {
    long r2 = row - (long)NB * NT;
    if (r2 >= (long)NB * NJ) return;
    src = U + r2 * ND; w = w_u; dst = su + r2;
  }
  float acc = 0.0f;
#pragma unroll
  for (int d = 0; d < ND / 32; ++d) acc += src[d * 32 + lane] * w[d * 32 + lane];
#pragma unroll
  for (int off = 16; off >= 1; off >>= 1) acc += __shfl_xor(acc, off, 32);
  if (lane == 0) *dst = acc;
}

// ---------------- k1: S via WMMA, softmax over J, C2Q via WMMA -------------
__global__ __launch_bounds__(256) void k1_attn(
    const float* __restrict__ H, const float* __restrict__ U,
    const float* __restrict__ w_hu, const float* __restrict__ sh,
    const float* __restrict__ su, float* __restrict__ mrow,
    float* __restrict__ G) {
  alignas(16) __shared__ float lds[LDS_FLOATS];
  const int b  = blockIdx.x >> 3;
  const int t0 = (blockIdx.x & 7) * TB;
  const int tid = threadIdx.x;
  const int wave = tid >> 5, lane = tid & 31;
  const int lh = lane >> 4;          // lane half (selects K pair / M group)
  const int ll = lane & 15;
  const float* Hb = H + ((long)(b * NT + t0)) * ND;
  const float* Ub = U + (long)b * NJ * ND;

  if (tid < NJ) lds[SU_OFF + tid] = su[b * NJ + tid];

  v8f accS[8];
#pragma unroll
  for (int c = 0; c < 8; ++c) accS[c] = zero8();

  // ---- Phase 1: S[16 rows x 128 cols] per wave, K over D in chunks of 64 ----
  for (int ch = 0; ch < 4; ++ch) {
    const int d0 = ch * KC;
    __syncthreads();
    for (int i = tid; i < NJ * (KC / 4); i += 256) {
      const int row = i >> 4, c4 = (i & 15) << 2;
      const float4 uv = *(const float4*)(Ub + (long)row * ND + d0 + c4);
      float* p = &lds[US_OFF + row * US_STRIDE + c4];
      p[0] = uv.x; p[1] = uv.y; p[2] = uv.z; p[3] = uv.w;
      const float4 hv = *(const float4*)(Hb + (long)row * ND + d0 + c4);
      const float4 wv = *(const float4*)(w_hu + d0 + c4);
      float* q = &lds[HS_OFF + row * US_STRIDE + c4];
      q[0] = hv.x * wv.x; q[1] = hv.y * wv.y; q[2] = hv.z * wv.z; q[3] = hv.w * wv.w;
    }
    __syncthreads();
    const int arow = wave * 16 + ll;     // M row for A fragment
    for (int kk = 0; kk < KC / 4; ++kk) {
      const int k0 = kk * 4 + lh * 2;
      v2f a = *(const v2f*)&lds[HS_OFF + arow * US_STRIDE + k0];
#pragma unroll
      for (int c = 0; c < 8; ++c) {
        v2f bb = *(const v2f*)&lds[US_OFF + (c * 16 + ll) * US_STRIDE + k0];
        accS[c] = __builtin_amdgcn_wmma_f32_16x16x4_f32(
            false, a, false, bb, (short)0, accS[c], false, false);
      }
    }
  }

  // ---- softmax over J (sh cancels; su added per column) ----
  float mx[8], rs[8];
#pragma unroll
  for (int v = 0; v < 8; ++v) mx[v] = -3.0e38f;
#pragma unroll
  for (int c = 0; c < 8; ++c) {
    const float suv = lds[SU_OFF + c * 16 + ll];
#pragma unroll
    for (int v = 0; v < 8; ++v) {
      accS[c][v] += suv;
      mx[v] = fmaxf(mx[v], accS[c][v]);
    }
  }
#pragma unroll
  for (int v = 0; v < 8; ++v) {
#pragma unroll
    for (int off = 8; off >= 1; off >>= 1)
      mx[v] = fmaxf(mx[v], __shfl_xor(mx[v], off, 32));
    rs[v] = 0.0f;
  }
#pragma unroll
  for (int c = 0; c < 8; ++c)
#pragma unroll
    for (int v = 0; v < 8; ++v) {
      const float e = __expf(accS[c][v] - mx[v]);
      accS[c][v] = e;
      rs[v] += e;
    }
#pragma unroll
  for (int v = 0; v < 8; ++v) {
#pragma unroll
    for (int off = 8; off >= 1; off >>= 1) rs[v] += __shfl_xor(rs[v], off, 32);
    rs[v] = 1.0f / rs[v];
  }
  if (ll == 0) {
#pragma unroll
    for (int v = 0; v < 8; ++v) {
      const int trow = t0 + wave * 16 + lh * 8 + v;
      mrow[b * NT + trow] = mx[v] + sh[b * NT + trow];  // add sh back for Q2C path
    }
  }

  // ---- store normalized A to LDS (row-major per wave, padded stride) ----
  __syncthreads();   // everyone done reading the H-chunk region
#pragma unroll
  for (int c = 0; c < 8; ++c)
#pragma unroll
    for (int v = 0; v < 8; ++v)
      lds[AS_OFF + wave * AS_WAVE + (lh * 8 + v) * AS_STRIDE + c * 16 + ll] =
          accS[c][v] * rs[v];
  __syncthreads();

  // ---- Phase 2: C2Q = A(16x128) * U(128x256), d-chunked; fuse G writes ----
  const float* As = &lds[AS_OFF + wave * AS_WAVE];
  for (int ch = 0; ch < 4; ++ch) {
    const int d0 = ch * KC;
    __syncthreads();
    for (int i = tid; i < NJ * (KC / 4); i += 256) {
      const int row = i >> 4, c4 = (i & 15) << 2;
      const float4 uv = *(const float4*)(Ub + (long)row * ND + d0 + c4);
      float* p = &lds[US_OFF + row * US_STRIDE + c4];
      p[0] = uv.x; p[1] = uv.y; p[2] = uv.z; p[3] = uv.w;
    }
    __syncthreads();
    v8f accC[4];
#pragma unroll
    for (int c = 0; c < 4; ++c) accC[c] = zero8();
    for (int kk = 0; kk < NJ / 4; ++kk) {
      const int k0 = kk * 4 + lh * 2;      // K = question index j
      v2f a = *(const v2f*)&As[ll * AS_STRIDE + k0];
#pragma unroll
      for (int c = 0; c < 4; ++c) {
        v2f bb;
        bb.x = lds[US_OFF + k0 * US_STRIDE + c * 16 + ll];
        bb.y = lds[US_OFF + (k0 + 1) * US_STRIDE + c * 16 + ll];
        accC[c] = __builtin_amdgcn_wmma_f32_16x16x4_f32(
            false, a, false, bb, (short)0, accC[c], false, false);
      }
    }
#pragma unroll
    for (int c = 0; c < 4; ++c)
#pragma unroll
      for (int v = 0; v < 8; ++v) {
        const int m = lh * 8 + v;
        const int trow = t0 + wave * 16 + m;
        const int d = d0 + c * 16 + ll;
        const long gbase = ((long)(b * NT + trow)) * (4 * ND);
        const float cv = accC[c][v];
        const float hv = Hb[(long)(wave * 16 + m) * ND + d];
        G[gbase + ND + d]     = cv;        // C2Q
        G[gbase + 2 * ND + d] = hv * cv;   // H * C2Q
      }
  }
}

// ---------------- k2: b_att = softmax_t(m) ; Q2C = b_att^T H ---------------
__global__ __launch_bounds__(1024) void k2_q2c(
    const float* __restrict__ mrow, const float* __restrict__ H,
    float* __restrict__ q2c) {
  __shared__ float batt[NT];
  __shared__ float red[32];
  __shared__ float part[4 * ND];
  const int b = blockIdx.x;
  const int tid = threadIdx.x;
  const int lane = tid & 31, wv = tid >> 5;
  const float v = mrow[b * NT + tid];
  float m = v;
#pragma unroll
  for (int off = 16; off >= 1; off >>= 1) m = fmaxf(m, __shfl_xor(m, off, 32));
  if (lane == 0) red[wv] = m;
  __syncthreads();
  if (wv == 0) {
    float t = red[lane];
#pragma unroll
    for (int off = 16; off >= 1; off >>= 1) t = fmaxf(t, __shfl_xor(t, off, 32));
    if (lane == 0) red[0] = t;
  }
  __syncthreads();
  m = red[0];
  const float e = __expf(v - m);
  float s = e;
#pragma unroll
  for (int off = 16; off >= 1; off >>= 1) s += __shfl_xor(s, off, 32);
  __syncthreads();
  if (lane == 0) red[wv] = s;
  __syncthreads();
  if (tid == 0) {
    float tot = 0.0f;
    for (int i = 0; i < 32; ++i) tot += red[i];
    red[0] = tot;
  }
  __syncthreads();
  batt[tid] = e / red[0];
  __syncthreads();
  const int d = tid & (ND - 1);
  const int g = tid >> 8;                 // 0..3 t-split
  const float* Hb = H + (long)b * NT * ND;
  float acc = 0.0f;
  for (int t = g; t < NT; t += 4) acc += batt[t] * Hb[(long)t * ND + d];
  part[g * ND + d] = acc;
  __syncthreads();
  if (g == 0)
    q2c[b * ND + d] = part[d] + part[ND + d] + part[2 * ND + d] + part[3 * ND + d];
}

// ---------------- k3: G[:,0:D]=H ; G[:,3D:4D]=H*Q2C ------------------------
__global__ __launch_bounds__(256) void k3_assemble(
    const float* __restrict__ H, const float* __restrict__ q2c,
    float* __restrict__ G) {
  const long row0 = (long)blockIdx.x * 4;
  const int d = threadIdx.x;
  const int b = (int)(row0 / NT);
  const float q = q2c[b * ND + d];
#pragma unroll
  for (int r = 0; r < 4; ++r) {
    const long t = row0 + r;
    const float h = H[t * ND + d];
    const long gbase = t * (4 * ND);
    G[gbase + d] = h;
    G[gbase + 3 * ND + d] = h * q;
  }
}

extern "C" void kernel_launch(void* const* d_in, const int* in_sizes, int n_in,
                              void* d_out, int out_size, void* d_ws, size_t ws_size,
                              hipStream_t stream) {
  (void)in_sizes; (void)n_in; (void)out_size; (void)ws_size;
  const float* H    = (const float*)d_in[0];
  const float* U    = (const float*)d_in[1];
  const float* w_h  = (const float*)d_in[2];
  const float* w_u  = (const float*)d_in[3];
  const float* w_hu = (const float*)d_in[4];
  float* G = (float*)d_out;

  float* sh   = (float*)d_ws;           // B*T
  float* su   = sh + NB * NT;           // B*J
  float* mrow = su + NB * NJ;           // B*T  (max_j S incl. sh)
  float* q2c  = mrow + NB * NT;         // B*D

  const int rows = NB * NT + NB * NJ;   // 73728, multiple of 8
  k0_rowdots<<<rows / 8, 256, 0, stream>>>(H, U, w_h, w_u, sh, su);
  k1_attn<<<NB * (NT / TB), 256, 0, stream>>>(H, U, w_hu, sh, su, mrow, G);
  k2_q2c<<<NB, 1024, 0, stream>>>(mrow, H, q2c);
  k3_assemble<<<NB * NT / 4, 256, 0, stream>>>(H, q2c, G);
}